// RationalInterpolationPoles1D_87531433493137
// MI455X (gfx1250) — compile-verified
//
#include <hip/hip_runtime.h>

#define N_NODES   1024
#define NEVAL_PTS 131072
#define NUM_POLES 8
#define EPS_F     1e-14f
#define PI_F      3.14159265358979323846f

typedef __attribute__((ext_vector_type(2))) float v2f;
typedef __attribute__((ext_vector_type(8))) float v8f;

// ---------------------------------------------------------------------------
// Kernel 1: barycentric weights with pole terms. One workgroup, 1024 threads.
// Writes to d_ws:  [0..1023] = values*w,  [1024..2047] = w,  [2048..3071] = nodes
// ---------------------------------------------------------------------------
__global__ __launch_bounds__(1024)
void bary_weights_kernel(const float* __restrict__ values,
                         const float* __restrict__ poles_real,
                         const float* __restrict__ poles_imag,
                         float* __restrict__ ws) {
  __shared__ float nodesS[N_NODES];
  __shared__ float red[N_NODES];
  __shared__ float w0abs;

  const int j = threadIdx.x;
  const float t = (float)j / (float)(N_NODES - 1);
  const float node = cosf(PI_F * t);   // domain (-1,1): _to_standard is identity
  nodesS[j] = node;
  __syncthreads();

  float logsum = 0.0f;
  int   neg = 0;
  for (int i = 0; i < N_NODES; ++i) {
    const float d = node - nodesS[i];
    if (i != j) logsum -= logf(fabsf(d));
    neg += (d < 0.0f) ? 1 : 0;
  }
  for (int m = 0; m < NUM_POLES; ++m) {
    const float dr = node - poles_real[m];
    const float ii = poles_imag[m];
    logsum += logf(dr * dr + ii * ii);
  }
  const float sign = (neg & 1) ? -1.0f : 1.0f;

  // max-reduce logsum across the workgroup
  red[j] = logsum;
  __syncthreads();
  for (int s = N_NODES / 2; s > 0; s >>= 1) {
    if (j < s) red[j] = fmaxf(red[j], red[j + s]);
    __syncthreads();
  }
  const float mx = red[0];

  float w = sign * expf(logsum - mx);
  if (j == 0) w0abs = fabsf(w);
  __syncthreads();
  w /= w0abs;

  ws[j]               = values[j] * w;   // vw
  ws[N_NODES + j]     = w;               // w
  ws[2 * N_NODES + j] = node;            // nodes (std == raw for this domain)
}

// ---------------------------------------------------------------------------
// Kernel 2: evaluation. 256 threads = 8 waves/block; each wave -> 16 columns.
// D(16x16) += A(16x4) x B(4x16) with V_WMMA_F32_16X16X4_F32:
//   A row 0 = vw[k..k+3], row 1 = w[k..k+3], rows 2..15 = 0 (zero LDS segment)
//   B[k][n] = 1/(x[n]-node[k])  (exact-hit override handled on the side)
// Result: c[0](lanes 0..15) = num, c[1](lanes 0..15) = den.
// ---------------------------------------------------------------------------
__global__ __launch_bounds__(256)
void bary_eval_kernel(const float* __restrict__ x_eval,
                      const float* __restrict__ ws,
                      float* __restrict__ out) {
  // ABp: [0..1023]=vw, [1024..2047]=w, [2048..3071]=0  (zero pad kills the
  // predicated A-loads: every lane does one unconditional ds_load_b64)
  __shared__ float ABp[3 * N_NODES];
  __shared__ float nodesS[N_NODES];

  const int tid = threadIdx.x;
  for (int i = tid; i < 3 * N_NODES; i += 256)
    ABp[i] = (i < 2 * N_NODES) ? ws[i] : 0.0f;
  for (int i = tid; i < N_NODES; i += 256)
    nodesS[i] = ws[2 * N_NODES + i];
  __syncthreads();

  const int wave  = tid >> 5;
  const int lane  = tid & 31;
  const int col   = lane & 15;          // eval column within tile / A-row index
  const int khalf = (lane >> 4) << 1;   // lanes 0-15 -> K{0,1}; 16-31 -> K{2,3}
  const int seg   = (col < 2) ? col : 2; // A segment: vw / w / zeros
  const int abase = seg * N_NODES + khalf;
  const int base  = (blockIdx.x * 8 + wave) * 16;

  const float x = x_eval[base + col];

  v8f c = {};
  float num2 = 0.0f, den2 = 0.0f;
  int   flag = 0;

  for (int kk = 0; kk < N_NODES; kk += 4) {
    const int k0 = kk + khalf;
    const int k1 = k0 + 1;
    const float d0 = x - nodesS[k0];
    const float d1 = x - nodesS[k1];
    const bool s0 = fabsf(d0) < EPS_F;
    const bool s1 = fabsf(d1) < EPS_F;

    // hardware rcp + one Newton step (~IEEE-divide accuracy, 3 VALU ops)
    float r0 = __builtin_amdgcn_rcpf(d0);
    r0 = fmaf(r0, fmaf(-d0, r0, 1.0f), r0);
    float r1 = __builtin_amdgcn_rcpf(d1);
    r1 = fmaf(r1, fmaf(-d1, r1, 1.0f), r1);

    const float b0 = s0 ? 1.0f : r0;
    const float b1 = s1 ? 1.0f : r1;

    if (s0 | s1) {                      // exact-hit columns: reference override
      flag = 1;
      if (s0) { num2 += ABp[k0]; den2 += ABp[N_NODES + k0]; }
      if (s1) { num2 += ABp[k1]; den2 += ABp[N_NODES + k1]; }
    }                                   // reconverged: EXEC all-ones at WMMA

    // A operand: unconditional contiguous pair load (vw | w | zero segment)
    const float a0 = ABp[abase + kk];
    const float a1 = ABp[abase + kk + 1];

    v2f A = {a0, a1};
    v2f B = {b0, b1};
    c = __builtin_amdgcn_wmma_f32_16x16x4_f32(
        /*neg_a=*/false, A, /*neg_b=*/false, B,
        /*c_mod=*/(short)0, c, /*reuse_a=*/false, /*reuse_b=*/false);
  }

  // combine exact-hit side accumulators across the two K-halves of the wave
  const float numO = __shfl_xor(num2, 16, 32);
  const float denO = __shfl_xor(den2, 16, 32);
  const int   flgO = __shfl_xor(flag, 16, 32);
  num2 += numO; den2 += denO; flag |= flgO;

  if (lane < 16) {
    const float num = c[0];   // D row M=0, N=lane
    const float den = c[1];   // D row M=1, N=lane
    out[base + lane] = flag ? (num2 / den2) : (num / den);
  }
}

// ---------------------------------------------------------------------------
extern "C" void kernel_launch(void* const* d_in, const int* in_sizes, int n_in,
                              void* d_out, int out_size, void* d_ws, size_t ws_size,
                              hipStream_t stream) {
  const float* x_eval     = (const float*)d_in[0];
  const float* values     = (const float*)d_in[1];
  const float* poles_real = (const float*)d_in[2];
  const float* poles_imag = (const float*)d_in[3];
  float* out = (float*)d_out;
  float* ws  = (float*)d_ws;   // needs 3*1024 floats = 12 KB

  bary_weights_kernel<<<1, N_NODES, 0, stream>>>(values, poles_real, poles_imag, ws);
  bary_eval_kernel<<<NEVAL_PTS / 128, 256, 0, stream>>>(x_eval, ws, out);
}